// NoiseGenerator_17179869184065
// MI455X (gfx1250) — compile-verified
//
#include <hip/hip_runtime.h>
#include <hip/hip_bf16.h>

// ---------------------------------------------------------------------------
// NoiseGenerator for MI455X (gfx1250, wave32).
//
// Blocked-scan implementation of two cascaded one-pole IIRs + exp envelope.
// Output traffic 98 MB -> ~4.2 us floor @ 23.3 TB/s; compute ~0.25 GFLOP is
// negligible, and WMMA is inapplicable (per-channel transition matrices mean
// no operand is shared across the lanes of a 16x16 tile).  Optimizations:
//   - 3-phase chunked scan: 188 chunks x 1024 channels -> thousands of waves
//     of parallelism over the serial time axis.
//   - Noise chunk staged in LDS (one coalesced 512B load per block), inner
//     loop reads via broadcast ds_load instead of per-iteration L2 round
//     trips inside the serial dependence chain.
//   - 2 channels per thread: two independent FMA recurrence chains per lane
//     (2x ILP against FMA latency) and 256B-per-wave b64 stores.
//   - global_prefetch hint for the noise stream; coalesced wave32 stores.
// ---------------------------------------------------------------------------

#define SR_INV      (1.0f / 48000.0f)
#define NSAMP       24000
#define BATCH       1024
#define EPSV        1e-4f
#define CHUNK       128                             // 2^7 samples per chunk
#define LOG2_CHUNK  7
#define NCHUNK      ((NSAMP + CHUNK - 1) / CHUNK)   // 188 (last chunk = 64)

// parameters layout: [5][BATCH] = attack, decay, f_lp, f_hp, gain
// buf layout:        [NCHUNK][2][BATCH] f32 (row 0: y1 state, row 1: y2 state)

// ---------------------------------------------------------------------------
// Phase 1: per-(channel-pair, chunk) zero-initial-state chunk result.
// ---------------------------------------------------------------------------
__global__ void __launch_bounds__(256)
k_partials(const float* __restrict__ params,
           const float* __restrict__ noise,
           float* __restrict__ buf) {
    __shared__ float s_noise[CHUNK];

    const int t  = blockIdx.x * blockDim.x + threadIdx.x;  // pair index, 0..511
    const int c  = blockIdx.y;                             // chunk
    const int n0 = c * CHUNK;
    const int n1 = min(NSAMP, n0 + CHUNK);

    // gfx1250 prefetch hint + one coalesced 512B stage of the noise window
    __builtin_prefetch(&noise[n0], 0, 1);
    if (threadIdx.x < CHUNK) {
        const int n = n0 + (int)threadIdx.x;
        s_noise[threadIdx.x] = (n < NSAMP) ? noise[n] : 0.0f;
    }
    __syncthreads();

    // two adjacent channels per thread -> two independent recurrence chains
    const float2 flp = *(const float2*)&params[2 * BATCH + 2 * t];
    const float2 fhp = *(const float2*)&params[3 * BATCH + 2 * t];
    const float a1A = flp.x, a1B = flp.y, c1A = 1.0f - flp.x, c1B = 1.0f - flp.y;
    const float a2A = fhp.x, a2B = fhp.y, c2A = 1.0f - fhp.x, c2B = 1.0f - fhp.y;

    float y1A = 0.0f, y2A = 0.0f, y1B = 0.0f, y2B = 0.0f;
    const int len = n1 - n0;
#pragma unroll 4
    for (int i = 0; i < len; ++i) {
        const float x = s_noise[i];            // LDS broadcast read
        y1A = fmaf(a1A, y1A, c1A * x);
        y1B = fmaf(a1B, y1B, c1B * x);
        y2A = fmaf(a2A, y2A, c2A * y1A);
        y2B = fmaf(a2B, y2B, c2B * y1B);
    }

    float2* buf2 = (float2*)buf;
    buf2[(2 * c + 0) * (BATCH / 2) + t] = make_float2(y1A, y1B);
    buf2[(2 * c + 1) * (BATCH / 2) + t] = make_float2(y2A, y2B);
}

// ---------------------------------------------------------------------------
// Phase 2: per-channel serial carry sweep across chunks (in-place).
// Chunk transition: s_c = M^CHUNK * s_{c-1} + d_c, lower-triangular
// M = [[a1,0],[(1-a2)a1, a2]];  squaring rule: (p,q,r) -> (p^2, q^2, r(p+q)).
// After this kernel, buf[c] holds the state at the END of chunk c.
// ---------------------------------------------------------------------------
__global__ void __launch_bounds__(256)
k_fixup(const float* __restrict__ params,
        float* __restrict__ buf) {
    const int b = blockIdx.x * blockDim.x + threadIdx.x;   // channel 0..1023
    const float a1 = params[2 * BATCH + b];
    const float a2 = params[3 * BATCH + b];
    const float c2 = 1.0f - a2;

    float p = a1, q = a2, r = c2 * a1;
#pragma unroll
    for (int i = 0; i < LOG2_CHUNK; ++i) {
        r = r * (p + q);
        p = p * p;
        q = q * q;
    }

    float s1 = 0.0f, s2 = 0.0f;
    for (int c = 0; c < NCHUNK; ++c) {
        const int i1 = (2 * c + 0) * BATCH + b;
        const int i2 = (2 * c + 1) * BATCH + b;
        const float d1 = buf[i1];
        const float d2 = buf[i2];
        const float t1 = fmaf(p, s1, d1);
        const float t2 = fmaf(r, s1, fmaf(q, s2, d2));  // uses old s1
        s1 = t1;
        s2 = t2;
        buf[i1] = s1;
        buf[i2] = s2;
    }
}

// ---------------------------------------------------------------------------
// Phase 3: per-(channel-pair, chunk) final pass seeded with carry[c-1];
// incremental envelope; 256B-per-wave coalesced b64 stores.
// ---------------------------------------------------------------------------
__global__ void __launch_bounds__(256)
k_emit(const float* __restrict__ params,
       const float* __restrict__ noise,
       const float* __restrict__ carry,
       float* __restrict__ out) {
    __shared__ float s_noise[CHUNK];

    const int t  = blockIdx.x * blockDim.x + threadIdx.x;  // pair index, 0..511
    const int c  = blockIdx.y;                             // chunk
    const int n0 = c * CHUNK;
    const int n1 = min(NSAMP, n0 + CHUNK);

    __builtin_prefetch(&noise[n0], 0, 1);
    if (threadIdx.x < CHUNK) {
        const int n = n0 + (int)threadIdx.x;
        s_noise[threadIdx.x] = (n < NSAMP) ? noise[n] : 0.0f;
    }
    __syncthreads();

    const float2 atk = *(const float2*)&params[0 * BATCH + 2 * t];
    const float2 dec = *(const float2*)&params[1 * BATCH + 2 * t];
    const float2 flp = *(const float2*)&params[2 * BATCH + 2 * t];
    const float2 fhp = *(const float2*)&params[3 * BATCH + 2 * t];
    const float2 gn  = *(const float2*)&params[4 * BATCH + 2 * t];

    const float a1A = flp.x, a1B = flp.y, c1A = 1.0f - flp.x, c1B = 1.0f - flp.y;
    const float a2A = fhp.x, a2B = fhp.y, c2A = 1.0f - fhp.x, c2B = 1.0f - fhp.y;

    const float iaA = 1.0f / (atk.x + EPSV), iaB = 1.0f / (atk.y + EPSV);
    const float idA = 1.0f / (dec.x + EPSV), idB = 1.0f / (dec.y + EPSV);

    // Envelope at chunk start (exact via expf), then one mul per sample.
    // ea = exp(-t/(attack+eps)); ed = exp(-t/(decay+eps)) * gain (gain folded).
    const float t0 = (float)n0 * SR_INV;
    float eaA = expf(-t0 * iaA), eaB = expf(-t0 * iaB);
    float edA = expf(-t0 * idA) * gn.x, edB = expf(-t0 * idB) * gn.y;
    const float raA = expf(-SR_INV * iaA), raB = expf(-SR_INV * iaB);
    const float rdA = expf(-SR_INV * idA), rdB = expf(-SR_INV * idB);

    float y1A, y2A, y1B, y2B;
    if (c == 0) {
        y1A = y2A = y1B = y2B = 0.0f;
    } else {
        const float2* carry2 = (const float2*)carry;
        const float2 s1 = carry2[(2 * (c - 1) + 0) * (BATCH / 2) + t];
        const float2 s2 = carry2[(2 * (c - 1) + 1) * (BATCH / 2) + t];
        y1A = s1.x; y1B = s1.y;
        y2A = s2.x; y2B = s2.y;
    }

    float2* out2 = (float2*)out;
    const int len = n1 - n0;
#pragma unroll 4
    for (int i = 0; i < len; ++i) {
        const float x = s_noise[i];            // LDS broadcast read
        y1A = fmaf(a1A, y1A, c1A * x);
        y1B = fmaf(a1B, y1B, c1B * x);
        y2A = fmaf(a2A, y2A, c2A * y1A);
        y2B = fmaf(a2B, y2B, c2B * y1B);
        const float envA = (1.0f - eaA) * edA;
        const float envB = (1.0f - eaB) * edB;
        out2[(n0 + i) * (BATCH / 2) + t] =
            make_float2((y1A - y2A) * envA, (y1B - y2B) * envB);
        eaA *= raA; eaB *= raB;
        edA *= rdA; edB *= rdB;
    }
}

// ---------------------------------------------------------------------------
extern "C" void kernel_launch(void* const* d_in, const int* in_sizes, int n_in,
                              void* d_out, int out_size, void* d_ws, size_t ws_size,
                              hipStream_t stream) {
    const float* params = (const float*)d_in[0];   // [5, 1024] f32
    const float* noise  = (const float*)d_in[1];   // [24000]   f32
    float*       out    = (float*)d_out;           // [24000, 1024] f32

    // Workspace: [NCHUNK][2][BATCH] f32 = 1.54 MB (partials, then in-place carries)
    float* buf = (float*)d_ws;

    const dim3 blk(256, 1, 1);                     // 8 wave32 waves per WG

    // Phase 1: 512 channel-pairs x 188 chunks = 376 blocks (3008 waves)
    k_partials<<<dim3(BATCH / 2 / 256, NCHUNK, 1), blk, 0, stream>>>(params, noise, buf);
    // Phase 2: 1024 channels, serial over 188 chunks
    k_fixup<<<dim3(BATCH / 256, 1, 1), blk, 0, stream>>>(params, buf);
    // Phase 3: same shape as phase 1, streams the 98MB output
    k_emit<<<dim3(BATCH / 2 / 256, NCHUNK, 1), blk, 0, stream>>>(params, noise, buf, out);
}